// CustomLoss_90537910600076
// MI455X (gfx1250) — compile-verified
//
#include <hip/hip_runtime.h>
#include <hip/hip_bf16.h>
#include <math.h>
#include <stdint.h>

// ---------------------------------------------------------------------------
// CustomLoss for MI455X (gfx1250).
// Memory-bound: 64MB streamed once -> ~2.8us roofline @ 23.3 TB/s.
// CDNA5 paths used: GLOBAL_LOAD_ASYNC_TO_LDS_B128 (ASYNCcnt) for LDS staging,
// s_wait_asynccnt, wave32-sized workgroups (8 waves/block), LDS dyadic
// window-max (O(log w) instead of O(w) VALU per element).
// ---------------------------------------------------------------------------

#define BB   2048          // batch
#define LL   4096          // row length
#define HH   2048          // half processed per LDS pass
#define FP   20            // front halo (>= distance-1 = 19, 16B-aligned data start)
#define BKP  20            // back halo
#define NP   (FP + HH + BKP)   // 2088 floats per LDS buffer
#define NT   256           // threads per block (8 x wave32)
#define EPT  (HH / NT)     // 8 elements per thread per half

#ifndef __has_builtin
#define __has_builtin(x) 0
#endif

#if __has_builtin(__builtin_amdgcn_global_load_async_to_lds_b128)
#define USE_ASYNC_BUILTIN 1
#else
#define USE_ASYNC_BUILTIN 0
#endif

typedef int v4i __attribute__((ext_vector_type(4)));
typedef __attribute__((address_space(1))) v4i* gptr_v4i;
typedef __attribute__((address_space(3))) v4i* lptr_v4i;

__device__ __forceinline__ void async_cp16(float* lds_dst, const float* gsrc) {
#if USE_ASYNC_BUILTIN
  __builtin_amdgcn_global_load_async_to_lds_b128(
      (gptr_v4i)(uintptr_t)gsrc,   // global src: AS1 int4*
      (lptr_v4i)lds_dst,           // LDS dst:   AS3 int4*
      /*offset=*/0, /*cpol=*/0);
#else
  // LDS byte offset: low 32 bits of the AS3 LDS address.
  unsigned lds_off = (unsigned)(uintptr_t)(__attribute__((address_space(3))) void*)lds_dst;
  unsigned long long ga = (unsigned long long)(uintptr_t)gsrc;
  asm volatile("global_load_async_to_lds_b128 %0, %1, off"
               :: "v"(lds_off), "v"(ga)
               : "memory");
#endif
}

__device__ __forceinline__ void wait_async_all() {
#if __has_builtin(__builtin_amdgcn_s_wait_asynccnt)
  __builtin_amdgcn_s_wait_asynccnt(0);
#else
  asm volatile("s_wait_asynccnt 0x0" ::: "memory");
#endif
}

__device__ __forceinline__ float rdmx(const float* b, int i) {
  return (i < NP) ? b[i] : -__builtin_inff();
}

// Dyadic (doubling) window max.
// On return: P[i] = max(X[i .. i+31]) (m5), Q[i] = max(X[i .. i+15]) (m4).
__device__ void winmax(const float* X, float* P, float* Q, int tid) {
  __syncthreads();
  for (int i = tid; i < NP; i += NT) P[i] = fmaxf(X[i], rdmx(X, i + 1));   // m1: win 2
  __syncthreads();
  for (int i = tid; i < NP; i += NT) Q[i] = fmaxf(P[i], rdmx(P, i + 2));   // m2: win 4
  __syncthreads();
  for (int i = tid; i < NP; i += NT) P[i] = fmaxf(Q[i], rdmx(Q, i + 4));   // m3: win 8
  __syncthreads();
  for (int i = tid; i < NP; i += NT) Q[i] = fmaxf(P[i], rdmx(P, i + 8));   // m4: win 16
  __syncthreads();
  for (int i = tid; i < NP; i += NT) P[i] = fmaxf(Q[i], rdmx(Q, i + 16));  // m5: win 32
  __syncthreads();
}

__device__ float blockReduce(float v, float* buf, int tid) {
  __syncthreads();
  buf[tid] = v;
  __syncthreads();
  for (int s = NT / 2; s > 0; s >>= 1) {
    if (tid < s) buf[tid] = buf[tid] + buf[tid + s];
    __syncthreads();
  }
  return buf[0];
}

__global__ __launch_bounds__(NT)
void loss_rows_kernel(const float* __restrict__ A, const float* __restrict__ R,
                      float* __restrict__ ws) {
  __shared__ __align__(16) float XI[NP];
  __shared__ __align__(16) float XR[NP];
  __shared__ __align__(16) float P[NP];
  __shared__ __align__(16) float Q[NP];

  const int tid = threadIdx.x;
  const int row = blockIdx.x;
  const float NEG = -__builtin_inff();
  const float* ga = A + (size_t)row * LL;
  const float* gr = R + (size_t)row * LL;

  float sumsq = 0.f, dotv = 0.f, na2 = 0.f, nb2 = 0.f;
  float nin = 0.f, nref = 0.f, p2p = 0.f;

  for (int h = 0; h < 2; ++h) {
    const int base = h * HH;
    __syncthreads();  // previous half's LDS reads complete before reload

    // Bulk: 2048 floats/signal = 512 float4 -> 2 async b128 per thread/signal.
    for (int c = 0; c < 2; ++c) {
      int v4 = c * NT + tid;
      async_cp16(&XI[FP + v4 * 4], ga + base + v4 * 4);
      async_cp16(&XR[FP + v4 * 4], gr + base + v4 * 4);
    }
    // Halos with -inf at row edges (matches reduce_window SAME/-inf padding).
    if (tid < FP) {
      int gj = base - FP + tid;
      XI[tid] = (gj >= 0) ? ga[gj] : NEG;
      XR[tid] = (gj >= 0) ? gr[gj] : NEG;
      int gj2 = base + HH + tid;
      XI[FP + HH + tid] = (gj2 < LL) ? ga[gj2] : NEG;
      XR[FP + HH + tid] = (gj2 < LL) ? gr[gj2] : NEG;
    }
    wait_async_all();
    __syncthreads();

    // ---------------- in_signal ----------------
    winmax(XI, P, Q, tid);  // P=m5(win32), Q=m4(win16)
    float pk[EPT];
    {
      const int j0 = tid * EPT;
      for (int k = 0; k < EPT; ++k) {
        const int j = base + j0 + k;   // global column
        const int i = FP + j0 + k;     // LDS index
        const float x = XI[i];
        const float r = XR[i];
        const float d = x - r;
        sumsq += d * d; dotv += x * r; na2 += x * x; nb2 += r * r;
        const bool lm = (j > 0) && (j < LL - 1) && (x > XI[i - 1]) && (x > XI[i + 1]);
        const float p20 = fmaxf(P[i - 19], P[i - 12]);  // max over [j-19, j+19]
        const float p10 = fmaxf(Q[i - 9],  Q[i - 6]);   // max over [j-9,  j+9]
        nin  += (lm && (x >= p20)) ? 1.f : 0.f;
        pk[k] = (lm && (x >= p10)) ? x : 0.f;
      }
    }
    __syncthreads();
    for (int k = 0; k < EPT; ++k) XI[FP + tid * EPT + k] = pk[k];  // XI := pk_in

    // ---------------- ref_signal ----------------
    winmax(XR, P, Q, tid);
    {
      const int j0 = tid * EPT;
      for (int k = 0; k < EPT; ++k) {
        const int j = base + j0 + k;
        const int i = FP + j0 + k;
        const float r = XR[i];
        const bool lm = (j > 0) && (j < LL - 1) && (r > XR[i - 1]) && (r > XR[i + 1]);
        const float p20 = fmaxf(P[i - 19], P[i - 12]);
        const float p10 = fmaxf(Q[i - 9],  Q[i - 6]);
        nref += (lm && (r >= p20)) ? 1.f : 0.f;
        const float pkr = (lm && (r >= p10)) ? r : 0.f;
        const float d = XI[i] - pkr;
        p2p += d * d;
      }
    }
  }

  // Deterministic per-row reductions (reuse P as scratch; NP >= NT).
  sumsq = blockReduce(sumsq, P, tid);
  dotv  = blockReduce(dotv,  P, tid);
  na2   = blockReduce(na2,   P, tid);
  nb2   = blockReduce(nb2,   P, tid);
  nin   = blockReduce(nin,   P, tid);
  nref  = blockReduce(nref,  P, tid);
  p2p   = blockReduce(p2p,   P, tid);

  if (tid == 0) {
    const float mse_i = sumsq * (1.0f / (float)LL);
    const float p2p_i = p2p   * (1.0f / (float)LL);
    const float cos_i = dotv / (sqrtf(na2) * sqrtf(nb2));
    const float custom = (nin != nref) ? (mse_i * 1.0f /*ALPHA*/)
                                       : (p2p_i * 0.5f /*BETA*/);
    float* o = ws + (size_t)row * 4;
    o[0] = sumsq;   // row sum of squared diff
    o[1] = cos_i;   // row cosine
    o[2] = custom;  // selected custom term
    o[3] = p2p_i;   // row p2p mean
  }
}

__global__ __launch_bounds__(NT)
void loss_final_kernel(const float* __restrict__ ws, float* __restrict__ out) {
  __shared__ float buf[NT];
  const int tid = threadIdx.x;
  float a = 0.f, b = 0.f, c = 0.f, d = 0.f;
  for (int r = tid; r < BB; r += NT) {
    const float* w = ws + (size_t)r * 4;
    a += w[0]; b += w[1]; c += w[2]; d += w[3];
  }
  a = blockReduce(a, buf, tid);
  b = blockReduce(b, buf, tid);
  c = blockReduce(c, buf, tid);
  d = blockReduce(d, buf, tid);
  if (tid == 0) {
    const float mse = a / (float)(BB) / (float)(LL);  // global mean sq diff
    out[0] = mse + c;          // total_loss
    out[1] = b / (float)BB;    // cosine_similarity (batch mean)
    out[2] = d;                // p2p_loss (sum of row means)
    out[3] = mse;              // mse_loss
  }
}

extern "C" void kernel_launch(void* const* d_in, const int* in_sizes, int n_in,
                              void* d_out, int out_size, void* d_ws, size_t ws_size,
                              hipStream_t stream) {
  const float* in_signal  = (const float*)d_in[0];
  const float* ref_signal = (const float*)d_in[1];
  float* out = (float*)d_out;
  float* ws  = (float*)d_ws;   // BB*4 floats = 32KB, fully written before read

  loss_rows_kernel<<<BB, NT, 0, stream>>>(in_signal, ref_signal, ws);
  loss_final_kernel<<<1, NT, 0, stream>>>(ws, out);
}